// VictorASIFractalLightModelAdvanced_30709016167060
// MI455X (gfx1250) — compile-verified
//
#include <hip/hip_runtime.h>
#include <hip/hip_bf16.h>
#include <stdint.h>

// ---------------------------------------------------------------------------
// VictorASIFractalLightModelAdvanced for MI455X (gfx1250, wave32, WMMA)
// fp32 end-to-end using V_WMMA_F32_16X16X4_F32 for every GEMM, with
// async global->LDS staging (GLOBAL_LOAD_ASYNC_TO_LDS_B128, ASYNCcnt).
// ---------------------------------------------------------------------------

#define MODEL_B   2
#define MODEL_L   2048
#define MODEL_D   1024
#define MODEL_T   (MODEL_B * MODEL_L)   // 4096 tokens
#define NLAYERS   2

#define GEMM_KC   32     // K-chunk staged in LDS
#define GEMM_PAD  36     // padded LDS row stride (floats); gcd(36,64)=4 -> conflict-free frag rows

typedef float v2f __attribute__((ext_vector_type(2)));
typedef float v8f __attribute__((ext_vector_type(8)));

// ---------------------------------------------------------------------------
// WMMA helper: D = A(16x4,f32) x B(4x16,f32) + C(16x16,f32)
// ---------------------------------------------------------------------------
__device__ __forceinline__ v8f wmma4(v2f a, v2f b, v8f c) {
  return __builtin_amdgcn_wmma_f32_16x16x4_f32(
      /*neg_a=*/false, a, /*neg_b=*/false, b,
      /*c_mod=*/(short)0, c, /*reuse_a=*/false, /*reuse_b=*/false);
}

// ---------------------------------------------------------------------------
// Async global -> LDS copy, 16B per lane (gfx1250, tracked by ASYNCcnt).
// ---------------------------------------------------------------------------
__device__ __forceinline__ void async_b128(float* lds_dst, const float* gsrc) {
  const unsigned lo = (unsigned)(uintptr_t)lds_dst;   // low 32 bits = LDS byte offset
  asm volatile("global_load_async_to_lds_b128 %0, %1, off"
               :: "v"(lo), "v"(gsrc) : "memory");
}

__device__ __forceinline__ void wait_async0() {
  asm volatile("s_wait_asynccnt 0x0" ::: "memory");
}

// ---------------------------------------------------------------------------
// Julia-set fractal embedding:  x[t,d] = emb_scale * sum_f feat[t,f]*proj_w[d,f]
// ---------------------------------------------------------------------------
__global__ __launch_bounds__(256) void julia_embed_kernel(
    const int* __restrict__ tok, const float* __restrict__ ctab,
    const float* __restrict__ projw, const float* __restrict__ escale,
    float* __restrict__ x) {
  const int t = blockIdx.x;
  const int tid = threadIdx.x;
  const int id = tok[t];
  const float cr = ctab[(size_t)id * 2 + 0];
  const float ci = ctab[(size_t)id * 2 + 1];
  float f[16];
  float zr = 0.f, zi = 0.f;
#pragma unroll
  for (int s = 0; s < 8; ++s) {
    float nr = zr * zr - zi * zi + cr;
    float ni = 2.f * zr * zi + ci;
    zr = nr; zi = ni;
    f[2 * s + 0] = zr;
    f[2 * s + 1] = zi;
  }
  const float es = escale[0];
#pragma unroll
  for (int j = 0; j < 4; ++j) {
    const int d = tid + j * 256;
    const float* w = projw + (size_t)d * 16;
    float acc = 0.f;
#pragma unroll
    for (int k = 0; k < 16; ++k) acc += f[k] * w[k];
    x[(size_t)t * MODEL_D + d] = acc * es;
  }
}

// ---------------------------------------------------------------------------
// Depthwise conv K=5, zero pad, per-channel bias.  y[t,d]
// ---------------------------------------------------------------------------
__global__ __launch_bounds__(256) void dwconv_kernel(
    const float* __restrict__ x, const float* __restrict__ w,
    const float* __restrict__ bias, float* __restrict__ y) {
  const int gid = blockIdx.x * 256 + threadIdx.x;
  const int t = gid >> 10;
  const int d = gid & 1023;
  const int b = t >> 11;
  const int l = t & 2047;
  const float* wd = w + (size_t)d * 5;
  float acc = bias[d];
#pragma unroll
  for (int k = 0; k < 5; ++k) {
    const int ll = l + k - 2;
    if (ll >= 0 && ll < MODEL_L)
      acc += x[((size_t)(b * MODEL_L + ll)) * MODEL_D + d] * wd[k];
  }
  y[(size_t)t * MODEL_D + d] = acc;
}

// ---------------------------------------------------------------------------
// Generic NT GEMM via WMMA f32 16x16x4 with double-buffered async LDS staging:
//   C[M,N] = A[M,K] * B[N,K]^T  (+ bias[n]) (+= C if accumulate)
// Block = 128 threads (4 waves), 64x64 C tile, 32x32 per wave.
// M,N multiples of 64; K multiple of 32 — true for all GEMMs in this model.
// ---------------------------------------------------------------------------
__device__ __forceinline__ void store_tile(
    float* __restrict__ C, const float* __restrict__ bias, int ldc,
    int m0, int n0, int half, int col, v8f c, int accumulate) {
#pragma unroll
  for (int e = 0; e < 8; ++e) {
    const size_t idx = (size_t)(m0 + 8 * half + e) * ldc + n0 + col;
    float v = c[e];
    if (bias) v += bias[n0 + col];
    if (accumulate) v += C[idx];
    C[idx] = v;
  }
}

// Issue async loads for one 64xKC tile pair (A and B) into buffers.
// Thread mapping: idx = tid + c*128 over 512 16B-chunks; 8 chunks per row.
__device__ __forceinline__ void issue_tile_loads(
    const float* __restrict__ A, const float* __restrict__ B,
    int lda, int ldb, int m0b, int n0b, int k0, int tid,
    float* bufA, float* bufB) {
#pragma unroll
  for (int c = 0; c < 4; ++c) {
    const int idx = tid + c * 128;
    const int row = idx >> 3;        // 0..63
    const int cir = idx & 7;         // 16B chunk within the 128B row
    const float* ga = A + (size_t)(m0b + row) * lda + k0 + cir * 4;
    const float* gb = B + (size_t)(n0b + row) * ldb + k0 + cir * 4;
    async_b128(bufA + row * GEMM_PAD + cir * 4, ga);
    async_b128(bufB + row * GEMM_PAD + cir * 4, gb);
  }
}

__global__ __launch_bounds__(128) void gemm_nt_kernel(
    const float* __restrict__ A, const float* __restrict__ B,
    float* __restrict__ C, const float* __restrict__ bias,
    int M, int N, int K, int lda, int ldb, int ldc, int accumulate) {
  __shared__ float tA[2][64 * GEMM_PAD];
  __shared__ float tB[2][64 * GEMM_PAD];

  const int tid = threadIdx.x;
  const int lane = tid & 31;
  const int wave = tid >> 5;
  const int half = lane >> 4;     // selects K-pair within frag / C row half
  const int r = lane & 15;        // frag row / C column
  const int m0b = blockIdx.y * 64;
  const int n0b = blockIdx.x * 64;
  const int wm = (wave >> 1) * 32;   // wave sub-tile inside 64x64
  const int wn = (wave & 1) * 32;

  v8f c00 = {}, c01 = {}, c10 = {}, c11 = {};

  const int nk = K / GEMM_KC;
  issue_tile_loads(A, B, lda, ldb, m0b, n0b, 0, tid, tA[0], tB[0]);

  int p = 0;
  for (int kc = 0; kc < nk; ++kc) {
    wait_async0();        // this wave's async writes for buffer p are in LDS
    __syncthreads();      // all waves' writes visible; all readers of p^1 done

    if (kc + 1 < nk)      // overlap next tile DMA with this tile's WMMAs
      issue_tile_loads(A, B, lda, ldb, m0b, n0b, (kc + 1) * GEMM_KC, tid,
                       tA[p ^ 1], tB[p ^ 1]);

    const float* sA = tA[p];
    const float* sB = tB[p];
#pragma unroll
    for (int kk = 0; kk < GEMM_KC; kk += 4) {
      const int ko = kk + 2 * half;
      const v2f a0 = *(const v2f*)&sA[(wm + r) * GEMM_PAD + ko];
      const v2f a1 = *(const v2f*)&sA[(wm + 16 + r) * GEMM_PAD + ko];
      const v2f b0 = *(const v2f*)&sB[(wn + r) * GEMM_PAD + ko];
      const v2f b1 = *(const v2f*)&sB[(wn + 16 + r) * GEMM_PAD + ko];
      c00 = wmma4(a0, b0, c00);
      c01 = wmma4(a0, b1, c01);
      c10 = wmma4(a1, b0, c10);
      c11 = wmma4(a1, b1, c11);
    }
    p ^= 1;
  }

  store_tile(C, bias, ldc, m0b + wm,      n0b + wn,      half, r, c00, accumulate);
  store_tile(C, bias, ldc, m0b + wm,      n0b + wn + 16, half, r, c01, accumulate);
  store_tile(C, bias, ldc, m0b + wm + 16, n0b + wn,      half, r, c10, accumulate);
  store_tile(C, bias, ldc, m0b + wm + 16, n0b + wn + 16, half, r, c11, accumulate);
}

// ---------------------------------------------------------------------------
// GLU gate + residual + LayerNorm (conv block epilogue).
// ---------------------------------------------------------------------------
__global__ __launch_bounds__(256) void glu_ln_kernel(
    const float* __restrict__ xin, const float* __restrict__ z,
    const float* __restrict__ gbias, const float* __restrict__ lng,
    const float* __restrict__ lnb, float* __restrict__ out) {
  const int t = blockIdx.x;
  const int tid = threadIdx.x;
  __shared__ float red[256];
  float h[4];
  float partial = 0.f;
#pragma unroll
  for (int j = 0; j < 4; ++j) {
    const int d = tid + j * 256;
    const float sig = z[(size_t)t * 2048 + d];
    const float gate = z[(size_t)t * 2048 + 1024 + d];
    const float gv = sig * (1.f / (1.f + __expf(-(gate + gbias[d]))));
    const float v = xin[(size_t)t * MODEL_D + d] + gv;
    h[j] = v;
    partial += v;
  }
  red[tid] = partial; __syncthreads();
  for (int s = 128; s > 0; s >>= 1) {
    if (tid < s) red[tid] += red[tid + s];
    __syncthreads();
  }
  const float mean = red[0] * (1.f / 1024.f);
  __syncthreads();
  partial = 0.f;
#pragma unroll
  for (int j = 0; j < 4; ++j) { const float dv = h[j] - mean; partial += dv * dv; }
  red[tid] = partial; __syncthreads();
  for (int s = 128; s > 0; s >>= 1) {
    if (tid < s) red[tid] += red[tid + s];
    __syncthreads();
  }
  const float var = red[0] * (1.f / 1024.f);
  const float rstd = rsqrtf(var + 1e-5f);
#pragma unroll
  for (int j = 0; j < 4; ++j) {
    const int d = tid + j * 256;
    out[(size_t)t * MODEL_D + d] = (h[j] - mean) * rstd * lng[d] + lnb[d];
  }
}

// ---------------------------------------------------------------------------
// Residual add + LayerNorm (attention block epilogue).
// ---------------------------------------------------------------------------
__global__ __launch_bounds__(256) void add_ln_kernel(
    const float* __restrict__ a, const float* __restrict__ b,
    const float* __restrict__ lng, const float* __restrict__ lnb,
    float* __restrict__ out) {
  const int t = blockIdx.x;
  const int tid = threadIdx.x;
  __shared__ float red[256];
  float h[4];
  float partial = 0.f;
#pragma unroll
  for (int j = 0; j < 4; ++j) {
    const int d = tid + j * 256;
    const float v = a[(size_t)t * MODEL_D + d] + b[(size_t)t * MODEL_D + d];
    h[j] = v;
    partial += v;
  }
  red[tid] = partial; __syncthreads();
  for (int s = 128; s > 0; s >>= 1) {
    if (tid < s) red[tid] += red[tid + s];
    __syncthreads();
  }
  const float mean = red[0] * (1.f / 1024.f);
  __syncthreads();
  partial = 0.f;
#pragma unroll
  for (int j = 0; j < 4; ++j) { const float dv = h[j] - mean; partial += dv * dv; }
  red[tid] = partial; __syncthreads();
  for (int s = 128; s > 0; s >>= 1) {
    if (tid < s) red[tid] += red[tid + s];
    __syncthreads();
  }
  const float var = red[0] * (1.f / 1024.f);
  const float rstd = rsqrtf(var + 1e-5f);
#pragma unroll
  for (int j = 0; j < 4; ++j) {
    const int d = tid + j * 256;
    out[(size_t)t * MODEL_D + d] = (h[j] - mean) * rstd * lng[d] + lnb[d];
  }
}

// ---------------------------------------------------------------------------
// Repack kv (B,L,2048) into k-pack (B,2,L,512) and v-transposed (B,2,512,L).
// kv[b,l, head*128 + {0:k,64:v} + hd], head = G*8 + h, c = h*64+hd.
// ---------------------------------------------------------------------------
__global__ __launch_bounds__(256) void repack_kv_kernel(
    const float* __restrict__ kv, float* __restrict__ kp, float* __restrict__ vt) {
  const int idx = blockIdx.x;           // (b*2+G)*L + l
  const int l = idx & 2047;
  const int G = (idx >> 11) & 1;
  const int b = idx >> 12;
#pragma unroll
  for (int j = 0; j < 2; ++j) {
    const int c = threadIdx.x + j * 256;
    const int h = c >> 6;
    const int hd = c & 63;
    const size_t src = ((size_t)(b * MODEL_L + l)) * 2048 + (size_t)(G * 8 + h) * 128 + hd;
    kp[(size_t)idx * 512 + c] = kv[src];
    vt[((size_t)(b * 2 + G) * 512 + c) * MODEL_L + l] = kv[src + 64];
  }
}

// ---------------------------------------------------------------------------
// Row softmax over L=2048 with score scale 0.125 * FRACTAL^G, in place.
// ---------------------------------------------------------------------------
__global__ __launch_bounds__(256) void softmax_kernel(float* __restrict__ s) {
  const int bid = blockIdx.x;
  const int tid = threadIdx.x;
  const int G = (bid >> 11) & 1;
  const float scale = 0.125f * (G ? 2.f : 1.f);
  float* row = s + (size_t)bid * 2048;
  __shared__ float red[256];
  float v[8];
  float mx = -3.0e38f;
#pragma unroll
  for (int j = 0; j < 8; ++j) {
    v[j] = row[tid + j * 256] * scale;
    mx = fmaxf(mx, v[j]);
  }
  red[tid] = mx; __syncthreads();
  for (int st = 128; st > 0; st >>= 1) {
    if (tid < st) red[tid] = fmaxf(red[tid], red[tid + st]);
    __syncthreads();
  }
  mx = red[0];
  __syncthreads();
  float sum = 0.f;
#pragma unroll
  for (int j = 0; j < 8; ++j) { v[j] = __expf(v[j] - mx); sum += v[j]; }
  red[tid] = sum; __syncthreads();
  for (int st = 128; st > 0; st >>= 1) {
    if (tid < st) red[tid] += red[tid + st];
    __syncthreads();
  }
  const float inv = 1.f / red[0];
#pragma unroll
  for (int j = 0; j < 8; ++j) row[tid + j * 256] = v[j] * inv;
}

// ---------------------------------------------------------------------------
// Host orchestration.
// ---------------------------------------------------------------------------
extern "C" void kernel_launch(void* const* d_in, const int* in_sizes, int n_in,
                              void* d_out, int out_size, void* d_ws, size_t ws_size,
                              hipStream_t stream) {
  (void)in_sizes; (void)n_in; (void)out_size; (void)ws_size;

  const int*   token_ids = (const int*)  d_in[0];
  const float* c_table   = (const float*)d_in[1];
  const float* proj_w    = (const float*)d_in[2];
  const float* emb_scale = (const float*)d_in[3];
  const float* dw_w      = (const float*)d_in[4];
  const float* dw_b      = (const float*)d_in[5];
  const float* pw_w      = (const float*)d_in[6];
  const float* pw_b      = (const float*)d_in[7];
  const float* gate_bias = (const float*)d_in[8];
  const float* conv_ln_g = (const float*)d_in[9];
  const float* conv_ln_b = (const float*)d_in[10];
  const float* q_w       = (const float*)d_in[11];
  const float* kv_w      = (const float*)d_in[12];
  const float* out_w     = (const float*)d_in[13];
  const float* attn_ln_g = (const float*)d_in[14];
  const float* attn_ln_b = (const float*)d_in[15];
  float* outp = (float*)d_out;

  // Workspace carve-up (floats).
  float* ws  = (float*)d_ws;
  const size_t TD  = (size_t)MODEL_T * MODEL_D;      // 4,194,304
  const size_t T2D = (size_t)MODEL_T * 2 * MODEL_D;  // 8,388,608
  float* x   = ws;                 // activations (layer input)
  float* xb  = x   + TD;           // conv-block output
  float* y   = xb  + TD;           // dwconv out / proj scratch
  float* z   = y   + TD;           // pointwise out (T x 2048)
  float* qb  = z   + T2D;          // q (T x 1024)
  float* kvb = qb  + TD;           // kv (T x 2048)
  float* kp  = kvb + T2D;          // k-pack  (B,2,L,512)
  float* vt  = kp  + TD;           // v-transposed (B,2,512,L)
  float* sb  = vt  + TD;           // scores (B,2,2,L,L) = 33,554,432 floats
  float* ao  = sb  + (size_t)8 * MODEL_L * MODEL_L;  // attn out pre-proj

  const dim3 blk_gemm(128);

  julia_embed_kernel<<<MODEL_T, 256, 0, stream>>>(token_ids, c_table, proj_w, emb_scale, x);

  for (int i = 0; i < NLAYERS; ++i) {
    // ---- conv block ----
    dwconv_kernel<<<(MODEL_T * MODEL_D) / 256, 256, 0, stream>>>(
        x, dw_w + (size_t)i * MODEL_D * 5, dw_b + (size_t)i * MODEL_D, y);

    gemm_nt_kernel<<<dim3(2048 / 64, MODEL_T / 64), blk_gemm, 0, stream>>>(
        y, pw_w + (size_t)i * 2048 * 1024, z, pw_b + (size_t)i * 2048,
        MODEL_T, 2048, 1024, 1024, 1024, 2048, 0);

    glu_ln_kernel<<<MODEL_T, 256, 0, stream>>>(
        x, z, gate_bias + (size_t)i * 1024,
        conv_ln_g + (size_t)i * 1024, conv_ln_b + (size_t)i * 1024, xb);

    // ---- attention block ----
    gemm_nt_kernel<<<dim3(1024 / 64, MODEL_T / 64), blk_gemm, 0, stream>>>(
        xb, q_w + (size_t)i * 1024 * 1024, qb, nullptr,
        MODEL_T, 1024, 1024, 1024, 1024, 1024, 0);

    gemm_nt_kernel<<<dim3(2048 / 64, MODEL_T / 64), blk_gemm, 0, stream>>>(
        xb, kv_w + (size_t)i * 2048 * 1024, kvb, nullptr,
        MODEL_T, 2048, 1024, 1024, 1024, 2048, 0);

    repack_kv_kernel<<<MODEL_B * 2 * MODEL_L, 256, 0, stream>>>(kvb, kp, vt);

    // scores: S[b,g,G] = Q_g K_G^T   (2048 x 2048, K=512)
    for (int b = 0; b < MODEL_B; ++b)
      for (int g = 0; g < 2; ++g)
        for (int G = 0; G < 2; ++G) {
          gemm_nt_kernel<<<dim3(2048 / 64, 2048 / 64), blk_gemm, 0, stream>>>(
              qb + (size_t)b * MODEL_L * 1024 + (size_t)g * 512,
              kp + (size_t)(b * 2 + G) * MODEL_L * 512,
              sb + (size_t)((b * 2 + g) * 2 + G) * MODEL_L * MODEL_L,
              nullptr, 2048, 2048, 512, 1024, 512, 2048, 0);
        }

    softmax_kernel<<<8 * MODEL_L, 256, 0, stream>>>(sb);

    // out_g = sum_G P[b,g,G] @ V_G   (2048 x 512, K=2048), accumulated over G
    for (int b = 0; b < MODEL_B; ++b)
      for (int g = 0; g < 2; ++g)
        for (int G = 0; G < 2; ++G) {
          gemm_nt_kernel<<<dim3(512 / 64, 2048 / 64), blk_gemm, 0, stream>>>(
              sb + (size_t)((b * 2 + g) * 2 + G) * MODEL_L * MODEL_L,
              vt + (size_t)(b * 2 + G) * 512 * MODEL_L,
              ao + (size_t)b * MODEL_L * 1024 + (size_t)g * 512,
              nullptr, 2048, 512, 2048, 2048, 2048, 1024, (G == 1) ? 1 : 0);
        }

    // out projection (reuse y as scratch)
    gemm_nt_kernel<<<dim3(1024 / 64, MODEL_T / 64), blk_gemm, 0, stream>>>(
        ao, out_w + (size_t)i * 1024 * 1024, y, nullptr,
        MODEL_T, 1024, 1024, 1024, 1024, 1024, 0);

    add_ln_kernel<<<MODEL_T, 256, 0, stream>>>(
        xb, y, attn_ln_g + (size_t)i * 1024, attn_ln_b + (size_t)i * 1024,
        (i == NLAYERS - 1) ? outp : x);
  }
}